// Model_89713276878902
// MI455X (gfx1250) — compile-verified
//
#include <hip/hip_runtime.h>

#define NN 100000      // nodes
#define NE 1600000     // edges
#define DIM 128
#define NG 512         // graphs
#define NL 3           // layers
#define OUTC (NL * DIM)
#define BN_EPS 1e-5f
#define ROWS_PER_BLOCK 128

typedef __attribute__((ext_vector_type(2))) float v2f;
typedef __attribute__((ext_vector_type(8))) float v8f;

// ---------------- utility kernels ----------------

__global__ void k_zero(float* __restrict__ p, int n) {
    int i = blockIdx.x * blockDim.x + threadIdx.x;
    if (i < n) p[i] = 0.f;
}

__global__ void k_init_deg(float* __restrict__ deg, int n) {
    int i = blockIdx.x * blockDim.x + threadIdx.x;
    if (i < n) deg[i] = 1.f;   // self-loop
}

__global__ void k_deg_edges(const int* __restrict__ dst, float* __restrict__ deg, int e) {
    int i = blockIdx.x * blockDim.x + threadIdx.x;
    if (i < e) atomicAdd(&deg[dst[i]], 1.f);
}

__global__ void k_dinv(float* __restrict__ deg, int n) {
    int i = blockIdx.x * blockDim.x + threadIdx.x;
    if (i < n) deg[i] = rsqrtf(fmaxf(deg[i], 1.f));
}

// 128x128 transpose: WT[n][k] = W[k][n]  (lets B fragments load as b64)
__global__ void __launch_bounds__(256)
k_transpose128(const float* __restrict__ W, float* __restrict__ WT) {
    __shared__ float t[16][17];                  // +1 pad: no bank conflicts
    int bx = blockIdx.x & 7, by = blockIdx.x >> 3;   // 8x8 grid of 16x16 tiles
    int tx = threadIdx.x & 15, ty = threadIdx.x >> 4;
    t[ty][tx] = W[(size_t)(by * 16 + ty) * DIM + bx * 16 + tx];
    __syncthreads();
    WT[(size_t)(bx * 16 + ty) * DIM + by * 16 + tx] = t[tx][ty];
}

// ---------------- WMMA fp32 GEMM: out[N,128] = A[N,128] @ W[128,128] ----------------
// One block = 16 output rows; 8 waves each own one 16-col tile of the 128 cols.
// A tile staged in LDS (ds_load_2addr_b64 fragments); B fragments are contiguous
// float2 loads from transposed W. K loop = 32 x V_WMMA_F32_16X16X4_F32.
// Safe in place (out == A): each block reads only its own 16 rows into LDS first.

__global__ void __launch_bounds__(256)
k_gemm_wmma(const float* __restrict__ A, const float* __restrict__ WT,
            float* __restrict__ out, int nrows) {
    __shared__ float tileA[16 * DIM];

    const int tid  = threadIdx.x;
    const int wave = tid >> 5;
    const int lane = tid & 31;
    const int row0 = blockIdx.x << 4;
    (void)nrows; // nrows % 16 == 0 here (100000 = 6250*16)

    // stage 16x128 floats = 512 float4, 2 per thread
    const float4* srcv = (const float4*)(A + (size_t)row0 * DIM);
    float4* dstv = (float4*)tileA;
    dstv[tid]       = srcv[tid];
    dstv[tid + 256] = srcv[tid + 256];
    __syncthreads();

    // fp32 WMMA 16x16x4 fragment mapping (wave32):
    //  A 16x4 : lane m = lane&15 holds row m; VGPR j -> K = 2*(lane>>4) + j
    //  B 4x16 : lane n = lane&15 holds col n; VGPR j -> K = 2*(lane>>4) + j
    //  C 16x16: VGPR r -> M = r + 8*(lane>>4), N = lane&15
    const int m  = lane & 15;
    const int kh = (lane >> 4) << 1;          // 0 or 2
    const int n  = (wave << 4) + (lane & 15); // this wave's output column

    const float2* wtrow = (const float2*)(WT + (size_t)n * DIM);

    v8f c = {};
#pragma unroll
    for (int ks = 0; ks < DIM; ks += 4) {
        v2f a, b;
        a.x = tileA[m * DIM + ks + kh];
        a.y = tileA[m * DIM + ks + kh + 1];
        float2 bb = wtrow[(ks + kh) >> 1];    // WT[n][ks+kh .. ks+kh+1] : one b64
        b.x = bb.x;
        b.y = bb.y;
        c = __builtin_amdgcn_wmma_f32_16x16x4_f32(
                /*neg_a=*/false, a, /*neg_b=*/false, b,
                /*c_mod=*/(short)0, c, /*reuse_a=*/false, /*reuse_b=*/false);
    }

    const int mo = (lane >> 4) << 3;          // 0 or 8
    float* orow = out + (size_t)(row0 + mo) * DIM + n;
#pragma unroll
    for (int r = 0; r < 8; ++r)
        orow[(size_t)r * DIM] = c[r];
}

// ---------------- aggregation ----------------

// out[n] = h[n] * dinv[n]^2  (self-loop term; also serves as the zero-init)
__global__ void k_selfloop(const float* __restrict__ h, const float* __restrict__ dinv,
                           float* __restrict__ out) {
    int i = blockIdx.x * blockDim.x + threadIdx.x;  // over NN*32 float4 chunks
    int node = i >> 5, q = i & 31;
    if (node < NN) {
        float w = dinv[node]; w *= w;
        float4 v = ((const float4*)(h + (size_t)node * DIM))[q];
        float4 r; r.x = v.x * w; r.y = v.y * w; r.z = v.z * w; r.w = v.w * w;
        ((float4*)(out + (size_t)node * DIM))[q] = r;
    }
}

// one wave per edge, lane q handles features [4q,4q+4): gather h[src], scale, scatter-add
__global__ void k_edge_agg(const int* __restrict__ src, const int* __restrict__ dst,
                           const float* __restrict__ dinv, const float* __restrict__ h,
                           float* __restrict__ out) {
    int i = blockIdx.x * blockDim.x + threadIdx.x;  // over NE*32
    int e = i >> 5, q = i & 31;
    if (e < NE) {
        int s = src[e], d = dst[e];
        float w = dinv[s] * dinv[d];
        float4 v = ((const float4*)(h + (size_t)s * DIM))[q];
        float* o = out + (size_t)d * DIM + (q << 2);
        atomicAdd(o + 0, v.x * w);
        atomicAdd(o + 1, v.y * w);
        atomicAdd(o + 2, v.z * w);
        atomicAdd(o + 3, v.w * w);
    }
}

// ---------------- epilogue: bias + relu + BN stats ----------------

// thread = channel; block handles ROWS_PER_BLOCK rows; h updated in place
__global__ void __launch_bounds__(DIM)
k_bias_relu_stats(float* __restrict__ h, const float* __restrict__ bias,
                  float* __restrict__ stats, int nrows) {
    int c  = threadIdx.x;
    int r0 = blockIdx.x * ROWS_PER_BLOCK;
    int r1 = min(r0 + ROWS_PER_BLOCK, nrows);
    float b = bias[c];
    float s = 0.f, s2 = 0.f;
    for (int r = r0; r < r1; ++r) {
        float v = h[(size_t)r * DIM + c] + b;
        v = fmaxf(v, 0.f);
        h[(size_t)r * DIM + c] = v;
        s += v; s2 += v * v;
    }
    atomicAdd(&stats[c], s);
    atomicAdd(&stats[DIM + c], s2);
}

__global__ void k_stats_finish(float* __restrict__ stats, float inv_n) {
    int c = threadIdx.x;
    float mu  = stats[c] * inv_n;
    float var = stats[DIM + c] * inv_n - mu * mu;   // biased variance (jnp.var)
    stats[2 * DIM + c] = mu;
    stats[3 * DIM + c] = rsqrtf(var + BN_EPS);
}

// normalize, write next-layer h, and pool per sorted graph id with run-length flush
__global__ void __launch_bounds__(DIM)
k_norm_pool(const float* __restrict__ h, const float* __restrict__ stats,
            const float* __restrict__ gamma, const float* __restrict__ beta,
            const int* __restrict__ batch, float* __restrict__ hnext,
            float* __restrict__ out, int layer, int nrows) {
    int c  = threadIdx.x;
    int r0 = blockIdx.x * ROWS_PER_BLOCK;
    int r1 = min(r0 + ROWS_PER_BLOCK, nrows);
    float mu = stats[2 * DIM + c], rstd = stats[3 * DIM + c];
    float ga = gamma[c], be = beta[c];
    int cur = -1; float acc = 0.f;
    for (int r = r0; r < r1; ++r) {
        float v = (h[(size_t)r * DIM + c] - mu) * rstd * ga + be;
        hnext[(size_t)r * DIM + c] = v;
        int g = batch[r];
        if (g != cur) {
            if (cur >= 0) atomicAdd(&out[(size_t)cur * OUTC + layer * DIM + c], acc);
            cur = g; acc = 0.f;
        }
        acc += v;
    }
    if (cur >= 0) atomicAdd(&out[(size_t)cur * OUTC + layer * DIM + c], acc);
}

// ---------------- launch ----------------

extern "C" void kernel_launch(void* const* d_in, const int* in_sizes, int n_in,
                              void* d_out, int out_size, void* d_ws, size_t ws_size,
                              hipStream_t stream) {
    (void)in_sizes; (void)n_in; (void)out_size; (void)ws_size;

    const float* x     = (const float*)d_in[0];
    const int*   ei    = (const int*)d_in[1];     // [2, NE] int32
    const int*   batch = (const int*)d_in[2];     // [NN]    int32
    const float *Wp[NL], *bp[NL], *gp[NL], *tp[NL];
    for (int i = 0; i < NL; ++i) {
        Wp[i] = (const float*)d_in[3 + 4 * i];
        bp[i] = (const float*)d_in[4 + 4 * i];
        gp[i] = (const float*)d_in[5 + 4 * i];
        tp[i] = (const float*)d_in[6 + 4 * i];
    }
    const int* esrc = ei;
    const int* edst = ei + NE;

    float* ws    = (float*)d_ws;
    float* dinv  = ws;                                   // NN floats
    float* stats = ws + NN;                              // 4*DIM floats
    float* wtb   = ws + NN + 4 * DIM;                    // DIM*DIM floats (W^T)
    size_t o     = ((size_t)NN + 4 * DIM + DIM * DIM + 255) & ~(size_t)255;
    float* bufA  = ws + o;                               // NN*DIM (gemm out / next h)
    float* bufB  = bufA + (size_t)NN * DIM;              // NN*DIM (aggregation)
    float* outp  = (float*)d_out;

    // init output + degree normalization
    k_zero     <<<(NG * OUTC + 255) / 256, 256, 0, stream>>>(outp, NG * OUTC);
    k_init_deg <<<(NN + 255) / 256,        256, 0, stream>>>(dinv, NN);
    k_deg_edges<<<(NE + 255) / 256,        256, 0, stream>>>(edst, dinv, NE);
    k_dinv     <<<(NN + 255) / 256,        256, 0, stream>>>(dinv, NN);

    for (int l = 0; l < NL; ++l) {
        const float* hin = (l == 0) ? x : bufA;
        // transpose layer weight so B fragments are contiguous b64 loads
        k_transpose128<<<64, 256, 0, stream>>>(Wp[l], wtb);
        // GEMM (in place for l>0: block-local LDS staging makes this safe)
        k_gemm_wmma<<<NN / 16, 256, 0, stream>>>(hin, wtb, bufA, NN);
        // self-loop term doubles as zero-init of bufB
        k_selfloop <<<(NN * 32 + 255) / 256, 256, 0, stream>>>(bufA, dinv, bufB);
        // edge scatter-add (L2-resident atomics)
        k_edge_agg <<<(NE * 32) / 256, 256, 0, stream>>>(esrc, edst, dinv, bufA, bufB);
        // BN stats over batch
        k_zero           <<<1, 256, 0, stream>>>(stats, 2 * DIM);
        k_bias_relu_stats<<<(NN + ROWS_PER_BLOCK - 1) / ROWS_PER_BLOCK, DIM, 0, stream>>>(
            bufB, bp[l], stats, NN);
        k_stats_finish   <<<1, DIM, 0, stream>>>(stats, 1.f / (float)NN);
        // normalize + emit next h + pooled output
        k_norm_pool      <<<(NN + ROWS_PER_BLOCK - 1) / ROWS_PER_BLOCK, DIM, 0, stream>>>(
            bufB, stats, gp[l], tp[l], batch, bufA, outp, l, NN);
    }
}